// BenGCN_1984274891423
// MI455X (gfx1250) — compile-verified
//
#include <hip/hip_runtime.h>
#include <math.h>

#define N_NODES 100000
#define N_EDGES 6400000
#define F_IN    128
#define HID     16
#define NCLS    2

typedef __attribute__((ext_vector_type(2))) float v2f;
typedef __attribute__((ext_vector_type(8))) float v8f;

// ---------------------------------------------------------------------------
// Degree / normalization precompute
// ---------------------------------------------------------------------------
__global__ void init_deg_kernel(float* __restrict__ deg) {
    int i = blockIdx.x * blockDim.x + threadIdx.x;
    if (i < N_NODES) deg[i] = 1.0f;   // self-loop weight 1
}

__global__ void accum_deg_kernel(const int* __restrict__ col,
                                 const float* __restrict__ ew,
                                 float* __restrict__ deg) {
    int e = blockIdx.x * blockDim.x + threadIdx.x;
    if (e < N_EDGES) atomicAdd(&deg[col[e]], ew[e]);
}

__global__ void dinv_kernel(float* __restrict__ deg) {
    int i = blockIdx.x * blockDim.x + threadIdx.x;
    if (i < N_NODES) {
        float d = deg[i];
        deg[i] = (d > 0.0f) ? rsqrtf(d) : 0.0f;
    }
}

__global__ void norm_kernel(const int* __restrict__ row,
                            const int* __restrict__ col,
                            const float* __restrict__ ew,
                            const float* __restrict__ dinv,
                            float* __restrict__ nrm) {
    long e = (long)blockIdx.x * blockDim.x + threadIdx.x;
    const long tot = (long)N_EDGES + N_NODES;
    if (e >= tot) return;
    if (e < N_EDGES) {
        nrm[e] = dinv[row[e]] * ew[e] * dinv[col[e]];
    } else {
        int i = (int)(e - N_EDGES);
        float d = dinv[i];
        nrm[e] = d * d;           // self loop: dinv[i]*1*dinv[i]
    }
}

// ---------------------------------------------------------------------------
// h1 = relu(x @ W_first^T + b_first)   [N,128] x [128,16] via WMMA f32 16x16x4
// One wave per 16-row tile, 8 waves per block.
// ---------------------------------------------------------------------------
__global__ __launch_bounds__(256)
void first_lin_wmma(const float* __restrict__ x,
                    const float* __restrict__ W,   // [HID, F_IN] row-major
                    const float* __restrict__ b,   // [HID]
                    float* __restrict__ h1) {      // [N, HID]
    int wave = threadIdx.x >> 5;
    int lane = threadIdx.x & 31;
    long tile = (long)blockIdx.x * (blockDim.x >> 5) + wave;
    long row0 = tile * 16;
    if (row0 >= N_NODES) return;                 // wave-uniform exit, EXEC stays full
    int half = lane >> 4;                        // 0: lanes 0-15, 1: lanes 16-31
    int l = lane & 15;

    long arow = row0 + l;
    if (arow >= N_NODES) arow = N_NODES - 1;     // clamp reads; stores guarded below
    const float* xr = x + arow * F_IN;

    v8f acc = {};
    #pragma unroll
    for (int k0 = 0; k0 < F_IN; k0 += 4) {
        int ka = k0 + 2 * half;
        v2f a;  a.x = xr[ka];                a.y = xr[ka + 1];
        // B[k][n] = W[n][k], n = l
        v2f bb; bb.x = W[l * F_IN + ka];     bb.y = W[l * F_IN + ka + 1];
        acc = __builtin_amdgcn_wmma_f32_16x16x4_f32(
            false, a, false, bb, (short)0, acc, false, false);
    }

    float bias = b[l];
    #pragma unroll
    for (int r = 0; r < 8; ++r) {
        long rr = row0 + r + half * 8;           // D: lanes16-31 hold M = r+8
        if (rr < N_NODES) {
            float v = acc[r] + bias;
            h1[rr * HID + l] = v > 0.0f ? v : 0.0f;
        }
    }
}

// ---------------------------------------------------------------------------
// m = h @ W^T   [N,16] x [16,16] via WMMA f32 16x16x4 (K = 16)
// ---------------------------------------------------------------------------
__global__ __launch_bounds__(256)
void hidden_wmma(const float* __restrict__ h,    // [N, HID]
                 const float* __restrict__ W,    // [HID, HID] row-major
                 float* __restrict__ m) {        // [N, HID]
    int wave = threadIdx.x >> 5;
    int lane = threadIdx.x & 31;
    long tile = (long)blockIdx.x * (blockDim.x >> 5) + wave;
    long row0 = tile * 16;
    if (row0 >= N_NODES) return;
    int half = lane >> 4;
    int l = lane & 15;

    long arow = row0 + l;
    if (arow >= N_NODES) arow = N_NODES - 1;
    const float* hr = h + arow * HID;

    v8f acc = {};
    #pragma unroll
    for (int k0 = 0; k0 < HID; k0 += 4) {
        int ka = k0 + 2 * half;
        v2f a;  a.x = hr[ka];              a.y = hr[ka + 1];
        v2f bb; bb.x = W[l * HID + ka];    bb.y = W[l * HID + ka + 1];
        acc = __builtin_amdgcn_wmma_f32_16x16x4_f32(
            false, a, false, bb, (short)0, acc, false, false);
    }

    #pragma unroll
    for (int r = 0; r < 8; ++r) {
        long rr = row0 + r + half * 8;
        if (rr < N_NODES) m[rr * HID + l] = acc[r];
    }
}

// ---------------------------------------------------------------------------
// agg[col[e]] += m[row[e]] * norm[e]  (one thread per edge, 16 f32 atomics)
// m (6.4 MB) lives in L2; row/col/norm stream from HBM once.
// ---------------------------------------------------------------------------
__global__ void scatter_agg(const float* __restrict__ m,
                            const int* __restrict__ row,
                            const int* __restrict__ col,
                            const float* __restrict__ nrm,
                            float* __restrict__ agg) {
    long e = (long)blockIdx.x * blockDim.x + threadIdx.x;
    const long tot = (long)N_EDGES + N_NODES;
    if (e >= tot) return;
    int r, c;
    if (e < N_EDGES) { r = row[e]; c = col[e]; }
    else             { r = c = (int)(e - N_EDGES); }      // self loop
    float nv = nrm[e];
    const float4* mr = (const float4*)(m + (long)r * HID);
    float* ag = agg + (long)c * HID;
    #pragma unroll
    for (int i = 0; i < 4; ++i) {
        float4 v = mr[i];
        atomicAdd(ag + 4 * i + 0, v.x * nv);
        atomicAdd(ag + 4 * i + 1, v.y * nv);
        atomicAdd(ag + 4 * i + 2, v.z * nv);
        atomicAdd(ag + 4 * i + 3, v.w * nv);
    }
}

__global__ void bias_relu_kernel(float* __restrict__ h,
                                 const float* __restrict__ b) {
    long i = (long)blockIdx.x * blockDim.x + threadIdx.x;
    if (i < (long)N_NODES * HID) {
        float v = h[i] + b[i & (HID - 1)];
        h[i] = v > 0.0f ? v : 0.0f;
    }
}

// ---------------------------------------------------------------------------
// logits = h @ W_out^T + b_out ; log_softmax (C = 2)
// ---------------------------------------------------------------------------
__global__ void out_logsoftmax(const float* __restrict__ h,
                               const float* __restrict__ Wo,  // [2, HID]
                               const float* __restrict__ bo,  // [2]
                               float* __restrict__ out) {     // [N, 2]
    long i = (long)blockIdx.x * blockDim.x + threadIdx.x;
    if (i >= N_NODES) return;
    const float* hr = h + i * HID;
    float l0 = bo[0], l1 = bo[1];
    #pragma unroll
    for (int k = 0; k < HID; ++k) {
        float v = hr[k];
        l0 += v * Wo[k];
        l1 += v * Wo[HID + k];
    }
    float mx = fmaxf(l0, l1);
    float lse = mx + logf(expf(l0 - mx) + expf(l1 - mx));
    out[i * 2 + 0] = l0 - lse;
    out[i * 2 + 1] = l1 - lse;
}

// ---------------------------------------------------------------------------
extern "C" void kernel_launch(void* const* d_in, const int* in_sizes, int n_in,
                              void* d_out, int out_size, void* d_ws, size_t ws_size,
                              hipStream_t stream) {
    const float* x       = (const float*)d_in[0];
    const int*   eidx    = (const int*)d_in[1];          // [2, E] flat
    const int*   row     = eidx;                         // sources
    const int*   col     = eidx + N_EDGES;               // targets
    const float* ew      = (const float*)d_in[2];
    const float* W_first = (const float*)d_in[3];
    const float* b_first = (const float*)d_in[4];
    const float* W_conv1 = (const float*)d_in[5];
    const float* b_conv1 = (const float*)d_in[6];
    const float* W_conv2 = (const float*)d_in[7];
    const float* b_conv2 = (const float*)d_in[8];
    const float* W_out   = (const float*)d_in[9];
    const float* b_out   = (const float*)d_in[10];
    float* out = (float*)d_out;

    float* ws   = (float*)d_ws;
    float* deg  = ws;                                    // N (becomes dinv)
    float* nrm  = deg + N_NODES;                         // E + N
    float* bufA = nrm + ((size_t)N_EDGES + N_NODES);     // N*HID
    float* bufB = bufA + (size_t)N_NODES * HID;          // N*HID
    float* bufM = bufB + (size_t)N_NODES * HID;          // N*HID

    const int T = 256;
    const long EN = (long)N_EDGES + N_NODES;
    int gN   = (N_NODES + T - 1) / T;
    int gE   = (N_EDGES + T - 1) / T;
    int gEN  = (int)((EN + T - 1) / T);
    int gNH  = (int)(((long)N_NODES * HID + T - 1) / T);
    int tiles  = (N_NODES + 15) / 16;
    int gWMMA  = (tiles + 7) / 8;                        // 8 waves / block

    // --- normalization coefficients (shared by both conv layers) ---
    init_deg_kernel<<<gN, T, 0, stream>>>(deg);
    accum_deg_kernel<<<gE, T, 0, stream>>>(col, ew, deg);
    dinv_kernel<<<gN, T, 0, stream>>>(deg);
    norm_kernel<<<gEN, T, 0, stream>>>(row, col, ew, deg, nrm);

    // --- h = relu(x @ W_first^T + b_first) -> bufA ---
    first_lin_wmma<<<gWMMA, T, 0, stream>>>(x, W_first, b_first, bufA);

    // --- conv1: m = bufA @ W1^T ; agg -> bufB ; relu(agg + b1) ---
    hidden_wmma<<<gWMMA, T, 0, stream>>>(bufA, W_conv1, bufM);
    hipMemsetAsync(bufB, 0, (size_t)N_NODES * HID * sizeof(float), stream);
    scatter_agg<<<gEN, T, 0, stream>>>(bufM, row, col, nrm, bufB);
    bias_relu_kernel<<<gNH, T, 0, stream>>>(bufB, b_conv1);

    // --- conv2: m = bufB @ W2^T ; agg -> bufA ; relu(agg + b2) ---
    hidden_wmma<<<gWMMA, T, 0, stream>>>(bufB, W_conv2, bufM);
    hipMemsetAsync(bufA, 0, (size_t)N_NODES * HID * sizeof(float), stream);
    scatter_agg<<<gEN, T, 0, stream>>>(bufM, row, col, nrm, bufA);
    bias_relu_kernel<<<gNH, T, 0, stream>>>(bufA, b_conv2);

    // --- output head + log_softmax ---
    out_logsoftmax<<<gN, T, 0, stream>>>(bufA, W_out, b_out, out);
}